// EnhancedGNNModel_10462540333260
// MI455X (gfx1250) — compile-verified
//
#include <hip/hip_runtime.h>
#include <hip/hip_bf16.h>

// ---------------------------------------------------------------------------
// GraphSAGE (4x SAGEConv, mean aggregation) for MI455X / gfx1250.
// CSR-gather aggregation (no float atomics), fp32 WMMA 16x16x4 for the
// 64->64 dense layers, scalar paths for the skinny 3->64 and 64->2 layers.
// ---------------------------------------------------------------------------

typedef __attribute__((ext_vector_type(2))) float v2f;
typedef __attribute__((ext_vector_type(8))) float v8f;

// ---------------- CSR construction ----------------

__global__ void k_degree(const long long* __restrict__ ei, long long E,
                         int* __restrict__ deg) {
  long long e = (long long)blockIdx.x * blockDim.x + threadIdx.x;
  if (e >= E) return;
  int dst = (int)ei[E + e];
  atomicAdd(&deg[dst], 1);
}

// Single-block exclusive scan of degrees -> row_ptr (+ cursor copy, 1/deg).
// 1024 threads = 32 wave32s; wave shuffle scan + LDS wave-sum scan.
__global__ void k_scan(const int* __restrict__ deg, int* __restrict__ row_ptr,
                       int* __restrict__ cursor, float* __restrict__ inv_deg,
                       int N) {
  __shared__ int ssum[32];
  __shared__ int scarry;
  int t = threadIdx.x;
  int lane = t & 31, w = t >> 5;
  if (t == 0) scarry = 0;
  __syncthreads();
  for (int base = 0; base < N; base += 1024) {
    int i = base + t;
    int v = (i < N) ? deg[i] : 0;
    int incl = v;
#pragma unroll
    for (int off = 1; off < 32; off <<= 1) {
      int xv = __shfl_up(incl, off, 32);
      if (lane >= off) incl += xv;
    }
    if (lane == 31) ssum[w] = incl;
    __syncthreads();
    if (w == 0) {
      int s = ssum[lane];
#pragma unroll
      for (int off = 1; off < 32; off <<= 1) {
        int xv = __shfl_up(s, off, 32);
        if (lane >= off) s += xv;
      }
      ssum[lane] = s;
    }
    __syncthreads();
    int waveoff = (w == 0) ? 0 : ssum[w - 1];
    int excl = scarry + waveoff + incl - v;
    if (i < N) {
      row_ptr[i] = excl;
      cursor[i] = excl;
      inv_deg[i] = 1.0f / (float)((v > 0) ? v : 1);
    }
    __syncthreads();
    if (t == 0) scarry += ssum[31];
    __syncthreads();
  }
  if (threadIdx.x == 0) row_ptr[N] = scarry;
}

__global__ void k_fill(const long long* __restrict__ ei, long long E,
                       int* __restrict__ cursor, int* __restrict__ col_src) {
  long long e = (long long)blockIdx.x * blockDim.x + threadIdx.x;
  if (e >= E) return;
  int src = (int)ei[e];
  int dst = (int)ei[E + e];
  int pos = atomicAdd(&cursor[dst], 1);
  col_src[pos] = src;
}

// ---------------- Mean aggregation (CSR gather) ----------------

// F=64: one wave per destination node, lane owns features [2*lane, 2*lane+1].
__global__ void k_aggregate64(const float* __restrict__ h,
                              const int* __restrict__ row_ptr,
                              const int* __restrict__ col_src,
                              const float* __restrict__ inv_deg,
                              float* __restrict__ agg, int N) {
  int node = (int)((blockIdx.x * blockDim.x + threadIdx.x) >> 5);
  int lane = threadIdx.x & 31;
  if (node >= N) return;
  int beg = row_ptr[node], end = row_ptr[node + 1];
  float ax = 0.f, ay = 0.f;
  for (int e = beg; e < end; ++e) {
    int s = col_src[e];
    const float2 v = *(const float2*)(h + (size_t)s * 64 + lane * 2);
    ax += v.x;
    ay += v.y;
  }
  float wgt = inv_deg[node];
  float2 r;
  r.x = ax * wgt;
  r.y = ay * wgt;
  *(float2*)(agg + (size_t)node * 64 + lane * 2) = r;
}

// F=3 (layer 1 input): lanes 0..2 carry the 3 features; agg stride 4 floats.
__global__ void k_aggregate3(const float* __restrict__ x,
                             const int* __restrict__ row_ptr,
                             const int* __restrict__ col_src,
                             const float* __restrict__ inv_deg,
                             float* __restrict__ agg4, int N) {
  int node = (int)((blockIdx.x * blockDim.x + threadIdx.x) >> 5);
  int lane = threadIdx.x & 31;
  if (node >= N) return;
  int beg = row_ptr[node], end = row_ptr[node + 1];
  float acc = 0.f;
  for (int e = beg; e < end; ++e) {
    int s = col_src[e];
    if (lane < 3) acc += x[(size_t)s * 3 + lane];
  }
  if (lane < 3) agg4[(size_t)node * 4 + lane] = acc * inv_deg[node];
}

// ---------------- Layer 1: (3 -> 64) + ReLU, scalar ----------------

__global__ void k_sage_small(const float* __restrict__ x,
                             const float* __restrict__ mean4,
                             const float* __restrict__ Wl,
                             const float* __restrict__ Wr,
                             const float* __restrict__ b,
                             float* __restrict__ out, int N) {
  int idx = blockIdx.x * blockDim.x + threadIdx.x;
  int node = idx >> 6, o = idx & 63;
  if (node >= N) return;
  float acc = b[o];
#pragma unroll
  for (int k = 0; k < 3; ++k)
    acc += mean4[(size_t)node * 4 + k] * Wl[k * 64 + o] +
           x[(size_t)node * 3 + k] * Wr[k * 64 + o];
  out[(size_t)node * 64 + o] = fmaxf(acc, 0.f);
}

// ---------------- Layers 2,3: (64 -> 64) + ReLU via fp32 WMMA ----------------
// out = relu(mean @ Wl + h @ Wr + b)
// One wave computes a 16-row x 64-col output stripe with 4 accumulators;
// K-loop 16 steps x (2 matrices x 4 col-tiles) = 128 v_wmma_f32_16x16x4_f32.
//
// Weights are staged in LDS *pre-paired* so each B fragment
// (W[k][col], W[k+1][col]) is a single aligned ds_load_b64:
//   sW[(k>>1)*128 + col*2 + (k&1)] = W[k*64 + col]
// A-row indices are clamped (not masked): output row m depends only on
// A row m, and rows >= N are never stored, so no zeroing/branching needed.

__global__ __launch_bounds__(256) void k_sage_wmma(
    const float* __restrict__ mean, const float* __restrict__ h,
    const float* __restrict__ Wl, const float* __restrict__ Wr,
    const float* __restrict__ bias, float* __restrict__ out, int N, int relu) {
  __shared__ float sWl[64 * 64];
  __shared__ float sWr[64 * 64];
  for (int i = threadIdx.x; i < 64 * 64; i += blockDim.x) {
    int k = i >> 6, col = i & 63;
    int d = (k >> 1) * 128 + col * 2 + (k & 1);
    sWl[d] = Wl[i];
    sWr[d] = Wr[i];
  }
  __syncthreads();

  int wave = threadIdx.x >> 5;  // 0..7
  int lane = threadIdx.x & 31;
  int row0 = ((int)blockIdx.x * 8 + wave) * 16;
  if (row0 >= N) return;  // wave-uniform exit (no syncs past this point)

  int m = lane & 15;   // A row / B,C column-within-tile
  int kh = lane >> 4;  // lane half selects K pair (A,B) and M half (C)
  int row = row0 + m;
  if (row >= N) row = N - 1;  // clamp: those A rows feed only unstored outputs
  const float* mrow = mean + (size_t)row * 64;
  const float* hrow = h + (size_t)row * 64;

  v8f acc0 = {}, acc1 = {}, acc2 = {}, acc3 = {};
  v8f acc[4] = {acc0, acc1, acc2, acc3};

  for (int k0 = 0; k0 < 64; k0 += 4) {
    int ka = k0 + 2 * kh;  // this lane-half's K base (A layout: K = vgpr + 2*half)
    v2f am = *(const v2f*)(mrow + ka);
    v2f ah = *(const v2f*)(hrow + ka);
    const int pb = (ka >> 1) * 128;  // paired-weight row base in LDS
#pragma unroll
    for (int t = 0; t < 4; ++t) {
      int col = t * 16 + m;  // B layout: N = lane&15, K = vgpr + 2*half
      v2f bl = *(const v2f*)(sWl + pb + col * 2);
      v2f br = *(const v2f*)(sWr + pb + col * 2);
      acc[t] = __builtin_amdgcn_wmma_f32_16x16x4_f32(false, am, false, bl,
                                                     (short)0, acc[t], false,
                                                     false);
      acc[t] = __builtin_amdgcn_wmma_f32_16x16x4_f32(false, ah, false, br,
                                                     (short)0, acc[t], false,
                                                     false);
    }
  }

  // C/D layout: VGPR v, lane half -> M = v + 8*half, N = lane&15
#pragma unroll
  for (int t = 0; t < 4; ++t) {
    int col = t * 16 + m;
    float bb = bias[col];
#pragma unroll
    for (int v = 0; v < 8; ++v) {
      int r = row0 + v + 8 * kh;
      if (r < N) {
        float val = acc[t][v] + bb;
        out[(size_t)r * 64 + col] = relu ? fmaxf(val, 0.f) : val;
      }
    }
  }
}

// ---------------- Layer 4: (64 -> 2) + log_softmax, scalar ----------------

__global__ void k_sage_out(const float* __restrict__ mean,
                           const float* __restrict__ h,
                           const float* __restrict__ Wl,
                           const float* __restrict__ Wr,
                           const float* __restrict__ b,
                           float* __restrict__ out, int N) {
  int i = blockIdx.x * blockDim.x + threadIdx.x;
  if (i >= N) return;
  const float* mr = mean + (size_t)i * 64;
  const float* hr = h + (size_t)i * 64;
  float z0 = b[0], z1 = b[1];
#pragma unroll 8
  for (int k = 0; k < 64; ++k) {
    float mv = mr[k], hv = hr[k];
    z0 += mv * Wl[k * 2 + 0] + hv * Wr[k * 2 + 0];
    z1 += mv * Wl[k * 2 + 1] + hv * Wr[k * 2 + 1];
  }
  float mx = fmaxf(z0, z1);
  float lse = mx + __logf(__expf(z0 - mx) + __expf(z1 - mx));
  out[(size_t)i * 2 + 0] = z0 - lse;
  out[(size_t)i * 2 + 1] = z1 - lse;
}

// ---------------- Host-side orchestration ----------------

extern "C" void kernel_launch(void* const* d_in, const int* in_sizes, int n_in,
                              void* d_out, int out_size, void* d_ws,
                              size_t ws_size, hipStream_t stream) {
  (void)n_in;
  (void)out_size;
  (void)ws_size;
  const float* x = (const float*)d_in[0];
  const long long* ei = (const long long*)d_in[1];
  const float* Wl1 = (const float*)d_in[2];
  const float* Wr1 = (const float*)d_in[3];
  const float* b1 = (const float*)d_in[4];
  const float* Wl2 = (const float*)d_in[5];
  const float* Wr2 = (const float*)d_in[6];
  const float* b2 = (const float*)d_in[7];
  const float* Wl3 = (const float*)d_in[8];
  const float* Wr3 = (const float*)d_in[9];
  const float* b3 = (const float*)d_in[10];
  const float* Wl4 = (const float*)d_in[11];
  const float* Wr4 = (const float*)d_in[12];
  const float* b4 = (const float*)d_in[13];

  const int N = in_sizes[0] / 3;
  const long long E = (long long)in_sizes[1] / 2;

  // Workspace carve-out (256 B aligned slices).
  char* ws = (char*)d_ws;
  size_t off = 0;
  auto alloc = [&](size_t bytes) -> void* {
    void* p = ws + off;
    off += (bytes + 255) & ~(size_t)255;
    return p;
  };
  int* deg = (int*)alloc((size_t)N * 4);
  int* row_ptr = (int*)alloc((size_t)(N + 1) * 4);
  int* cursor = (int*)alloc((size_t)N * 4);
  float* inv_deg = (float*)alloc((size_t)N * 4);
  int* col_src = (int*)alloc((size_t)E * 4);
  float* agg = (float*)alloc((size_t)N * 64 * 4);
  float* hA = (float*)alloc((size_t)N * 64 * 4);
  float* hB = (float*)alloc((size_t)N * 64 * 4);

  const int tpb = 256;
  const int gE = (int)((E + tpb - 1) / tpb);
  const int gWaveNode = (N * 32 + tpb - 1) / tpb;  // one wave32 per node
  const int gWmma = (N + 127) / 128;               // 8 waves x 16 rows / block

  // CSR build (every launch: no cross-call state allowed).
  hipMemsetAsync(deg, 0, (size_t)N * 4, stream);
  k_degree<<<gE, tpb, 0, stream>>>(ei, E, deg);
  k_scan<<<1, 1024, 0, stream>>>(deg, row_ptr, cursor, inv_deg, N);
  k_fill<<<gE, tpb, 0, stream>>>(ei, E, cursor, col_src);

  // Layer 1: 3 -> 64, ReLU
  k_aggregate3<<<gWaveNode, tpb, 0, stream>>>(x, row_ptr, col_src, inv_deg,
                                              agg, N);
  k_sage_small<<<(N * 64 + tpb - 1) / tpb, tpb, 0, stream>>>(x, agg, Wl1, Wr1,
                                                             b1, hA, N);
  // Layer 2: 64 -> 64, ReLU (WMMA)
  k_aggregate64<<<gWaveNode, tpb, 0, stream>>>(hA, row_ptr, col_src, inv_deg,
                                               agg, N);
  k_sage_wmma<<<gWmma, 256, 0, stream>>>(agg, hA, Wl2, Wr2, b2, hB, N, 1);
  // Layer 3: 64 -> 64, ReLU (WMMA)
  k_aggregate64<<<gWaveNode, tpb, 0, stream>>>(hB, row_ptr, col_src, inv_deg,
                                               agg, N);
  k_sage_wmma<<<gWmma, 256, 0, stream>>>(agg, hB, Wl3, Wr3, b3, hA, N, 1);
  // Layer 4: 64 -> 2 + log_softmax
  k_aggregate64<<<gWaveNode, tpb, 0, stream>>>(hA, row_ptr, col_src, inv_deg,
                                               agg, N);
  k_sage_out<<<(N + tpb - 1) / tpb, tpb, 0, stream>>>(agg, hA, Wl4, Wr4, b4,
                                                      (float*)d_out, N);
}